// PIS_75007308857881
// MI455X (gfx1250) — compile-verified
//
#include <hip/hip_runtime.h>
#include <stdint.h>

// Problem constants (match reference)
#define B_TOT     4096
#define N_FIELD   32
#define K_DIM     16
#define NI        496      // 32*31/2
#define NTILE     31       // NI / 16
#define D_VOCAB   100000
#define L_SAMPLES 4
#define L1_COEF   0.01f

typedef __attribute__((ext_vector_type(2))) float v2f;
typedef __attribute__((ext_vector_type(8))) float v8f;

// Counter-based hash RNG (stand-in for JAX threefry; same key structure:
// streams keyed by (sample, batch, pair) / (sample, field) / (sample, pair)).
__device__ __forceinline__ uint32_t mix3(uint32_t a, uint32_t b, uint32_t c) {
  uint32_t h = a * 0x9E3779B1u;
  h ^= (b + 0x85EBCA6Bu) * 0xC2B2AE35u;
  h ^= (c + 0x27D4EB2Fu) * 0x165667B1u;
  h ^= h >> 16; h *= 0x7FEB352Du;
  h ^= h >> 15; h *= 0x846CA68Bu;
  h ^= h >> 16;
  return h;
}
__device__ __forceinline__ float u01(uint32_t h) {
  return (float)h * 2.3283064365386963e-10f;  // [0,1)
}

// One wave32 per batch element. All intermediates (Vx tile, pi, mu/sigma,
// packed pair indices) live in LDS (~8.1 KB/workgroup -> full occupancy on
// the 320KB WGP LDS pool). The [NI,16]x[16,2] einsum runs on
// V_WMMA_F32_16X16X4_F32 (4 chained K-slices per 16-pair tile).
__global__ __launch_bounds__(32) void vfm_main(
    const int* __restrict__ x, const int* __restrict__ u,
    const int* __restrict__ rowi, const int* __restrict__ coli,
    const float* __restrict__ V, const float* __restrict__ pro,
    const float* __restrict__ w,
    const float* __restrict__ hmu, const float* __restrict__ hsig,
    float* __restrict__ out, float* __restrict__ klpart)
{
  __shared__ float    sVx[N_FIELD][K_DIM];  // 2048 B
  __shared__ float    sMu[NI];              // 1984 B
  __shared__ float    sSig[NI];             // 1984 B
  __shared__ float    sPi[N_FIELD];         //  128 B
  __shared__ uint32_t sRC[NI];              // 1984 B (row | col<<8)

  const int lane = threadIdx.x;             // 0..31 == field index
  const int b    = blockIdx.x;

  // ---- Gather phase -------------------------------------------------------
  const int xi = x[b * N_FIELD + lane];
  const int ub = u[b];

  const float4* vrow = (const float4*)(V + (size_t)xi * K_DIM);
  float4* vdst = (float4*)(&sVx[lane][0]);
  vdst[0] = vrow[0]; vdst[1] = vrow[1];
  vdst[2] = vrow[2]; vdst[3] = vrow[3];

  const float pi_i = pro[(size_t)ub * (size_t)D_VOCAB + (size_t)xi];
  sPi[lane] = pi_i;
  float wsum = w[xi];

  for (int p = lane; p < NI; p += 32)
    sRC[p] = (uint32_t)rowi[p] | ((uint32_t)coli[p] << 8);

  // wave32 butterfly reduce for wx (all lanes end with the sum)
  #pragma unroll
  for (int off = 16; off > 0; off >>= 1) wsum += __shfl_xor(wsum, off);

  // Per-sample stochastic field-selection masks: e1 depends only on (l, field),
  // so one ballot per sample gives the whole 32-bit s-vector for this batch row.
  uint32_t smask[L_SAMPLES];
  #pragma unroll
  for (int l = 0; l < L_SAMPLES; ++l) {
    const float e1 = u01(mix3(0xE1AA55u, (uint32_t)l, (uint32_t)lane));
    smask[l] = (uint32_t)__ballot(pi_i >= e1);
  }

  __syncthreads();

  // ---- WMMA phase: mu/sigma = inter[NI,16] x H[16,2] ----------------------
  const int n16 = lane & 15;   // A: M index / B,D: N index
  const int hi  = lane >> 4;   // 0: K=(0,1) half, 1: K=(2,3) half (A & B f32 layout)

  // B operand per K-slice: column 0 = h_mu, column 1 = h_sigma, rest zero.
  v2f bm[4];
  #pragma unroll
  for (int ks = 0; ks < 4; ++ks) {
    const int k0 = 4 * ks + 2 * hi;
    float bx = 0.f, by = 0.f;
    if (n16 == 0)      { bx = hmu[k0];  by = hmu[k0 + 1]; }
    else if (n16 == 1) { bx = hsig[k0]; by = hsig[k0 + 1]; }
    bm[ks].x = bx; bm[ks].y = by;
  }

  for (int t = 0; t < NTILE; ++t) {
    const int p = t * 16 + n16;
    const uint32_t rc = sRC[p];
    const int r = (int)(rc & 0xFFu), c = (int)((rc >> 8) & 0xFFu);
    v8f acc = {};
    #pragma unroll
    for (int ks = 0; ks < 4; ++ks) {
      const int k0 = 4 * ks + 2 * hi;
      v2f a;
      a.x = sVx[r][k0]     * sVx[c][k0];
      a.y = sVx[r][k0 + 1] * sVx[c][k0 + 1];
      // (neg_a, A, neg_b, B, c_mod, C, reuse_a, reuse_b)
      acc = __builtin_amdgcn_wmma_f32_16x16x4_f32(
          false, a, false, bm[ks], (short)0, acc, false, false);
    }
    // D layout: lanes 0-15 N=lane M=vgpr, lanes 16-31 N=lane-16 M=vgpr+8.
    // mu lives in N==0 (lanes 0,16), sigma in N==1 (lanes 1,17).
    if (n16 <= 1) {
      float* dst = (n16 == 0) ? sMu : sSig;
      const int mbase = t * 16 + hi * 8;
      #pragma unroll
      for (int j = 0; j < 8; ++j) dst[mbase + j] = acc[j];
    }
  }

  __syncthreads();

  // ---- Monte-Carlo + KL phase --------------------------------------------
  float second = 0.f, klp = 0.f;
  for (int p = lane; p < NI; p += 32) {
    const float mu = sMu[p], sg = sSig[p];
    const uint32_t rc = sRC[p];
    const int r = (int)(rc & 0xFFu), c = (int)((rc >> 8) & 0xFFu);
    const float Pi = sPi[r] * sPi[c];

    klp += 1.f + sg - mu * mu - __expf(sg);
    const float sdev = __expf(0.5f * sg);

    #pragma unroll
    for (int l = 0; l < L_SAMPLES; ++l) {
      // eps ~ N(0,1) via Box-Muller, keyed by (l, b, p)
      const uint32_t h1 = mix3(0x9E3700u + (uint32_t)l, (uint32_t)b, (uint32_t)p);
      const uint32_t h2 = mix3(0x51F15Eu + (uint32_t)l, (uint32_t)p, (uint32_t)b);
      const float u1v = ((float)h1 + 0.5f) * 2.3283064365386963e-10f; // (0,1)
      const float u2v = u01(h2);
      const float eps = sqrtf(-2.f * __logf(u1v)) * __cosf(6.28318530718f * u2v);
      const float z   = fmaf(eps, sdev, mu);

      // e2 depends only on (l, p) — shared across the batch, like reference.
      const float e2 = u01(mix3(0xE2BEEFu + (uint32_t)l, (uint32_t)p, 0x5EEDu));
      const int ssum = (int)((smask[l] >> r) & 1u)
                     + (int)((smask[l] >> c) & 1u)
                     + (Pi >= e2 ? 1 : 0);
      if (ssum >= 2) second += z;
    }
  }

  #pragma unroll
  for (int off = 16; off > 0; off >>= 1) {
    second += __shfl_xor(second, off);
    klp    += __shfl_xor(klp, off);
  }
  if (lane == 0) {
    out[b]    = second * (1.f / (float)L_SAMPLES) + wsum;
    klpart[b] = klp;   // per-batch KL partial (deterministic 2-pass reduce)
  }
}

// Single-block deterministic reduction of the 4096 KL partials, then add the
// uniform term w0 + L1 * (-0.5 * sum) to every output element.
__global__ __launch_bounds__(256) void vfm_finalize(
    const float* __restrict__ klpart, const float* __restrict__ w0,
    float* __restrict__ out)
{
  __shared__ float red[256];
  float s = 0.f;
  for (int i = (int)threadIdx.x; i < B_TOT; i += 256) s += klpart[i];
  red[threadIdx.x] = s;
  __syncthreads();
  for (int stride = 128; stride > 0; stride >>= 1) {
    if ((int)threadIdx.x < stride) red[threadIdx.x] += red[threadIdx.x + stride];
    __syncthreads();
  }
  const float add = w0[0] + L1_COEF * (-0.5f * red[0]);
  for (int i = (int)threadIdx.x; i < B_TOT; i += 256) out[i] += add;
}

extern "C" void kernel_launch(void* const* d_in, const int* in_sizes, int n_in,
                              void* d_out, int out_size, void* d_ws, size_t ws_size,
                              hipStream_t stream) {
  (void)in_sizes; (void)n_in; (void)out_size; (void)ws_size;
  const int*   x    = (const int*)d_in[0];
  const int*   u    = (const int*)d_in[1];
  const int*   row  = (const int*)d_in[2];
  const int*   col  = (const int*)d_in[3];
  const float* V    = (const float*)d_in[4];
  const float* pro  = (const float*)d_in[5];
  const float* w    = (const float*)d_in[6];
  const float* w0   = (const float*)d_in[7];
  const float* hmu  = (const float*)d_in[8];
  const float* hsig = (const float*)d_in[9];
  float* out    = (float*)d_out;
  float* klpart = (float*)d_ws;   // 4096 floats of scratch

  vfm_main<<<B_TOT, 32, 0, stream>>>(x, u, row, col, V, pro, w, hmu, hsig,
                                     out, klpart);
  vfm_finalize<<<1, 256, 0, stream>>>(klpart, w0, out);
}